// DetectionBoxLoss_36404142801048
// MI455X (gfx1250) — compile-verified
//
#include <hip/hip_runtime.h>

// ---------------------------------------------------------------------------
// CIoU detection-box loss, MI455X (gfx1250, wave32).
// Memory-bound streaming pass (78 MB @ 23.3 TB/s ~ 3.3 us floor) + tiny
// finalize. Grid centers cancel algebraically; atan difference folded into a
// single atan via atan(a)-atan(b) = atan((a-b)/(1+ab)) (a,b > 0 here), and
// atan itself is a 14-op branch-free minimax poly (|err| ~ 2e-6) instead of
// the ~30-op OCML atanf, since VALU time is the limiter past the HBM floor.
// Final 32-lane reduction done with V_WMMA_F32_16X16X4_F32 in full f32.
// ---------------------------------------------------------------------------

typedef float          v4f  __attribute__((ext_vector_type(4)));
typedef float          v2f  __attribute__((ext_vector_type(2)));
typedef float          v8f  __attribute__((ext_vector_type(8)));
typedef unsigned char  v4u8 __attribute__((ext_vector_type(4)));

#define BB        64
#define HH        192
#define WW        192
#define HW        (HH * WW)          // 36864
#define HW4       (HW / 4)           // 9216
#define NVEC      (BB * HW4)         // 589824 float4-groups, no tail
#define NBLK      512
#define NTHR      256
#define SS        8.0f               // STRIDE
#define EPSF      1e-7f
#define FOUR_PI2  0.405284734569351f // 4/pi^2
#define PI_2      1.57079632679489662f

__device__ __forceinline__ float fast_rcp(float x) {
    return __builtin_amdgcn_rcpf(x);
}

// Branch-free atan, full range, |abs err| ~ 2e-6:
//   t = min(|x|,1)/max(|x|,1);  p = odd minimax poly(t) on [0,1]
//   r = |x|>1 ? pi/2 - p : p;   result = copysign(r, x)
__device__ __forceinline__ float fast_atan(float x) {
    float ax = __builtin_fabsf(x);
    float hi = fmaxf(ax, 1.0f);
    float lo = fminf(ax, 1.0f);
    float t  = lo * fast_rcp(hi);
    float s  = t * t;
    float p  = -0.01172120f;
    p = fmaf(p, s,  0.05265332f);
    p = fmaf(p, s, -0.11643287f);
    p = fmaf(p, s,  0.19354346f);
    p = fmaf(p, s, -0.33262347f);
    p = fmaf(p, s,  0.99997726f);
    float r = t * p;
    r = (ax > 1.0f) ? (PI_2 - r) : r;
    return copysignf(r, x);
}

__device__ __forceinline__ void ciou_accum(float pl, float pt, float pr, float pb,
                                           float tl, float tt, float tr, float tb,
                                           float m, float& s0, float& s1) {
    // Box sizes (grid centers cancel everywhere)
    float w1 = SS * (pl + pr), h1 = SS * (pt + pb);
    float w2 = SS * (tl + tr), h2 = SS * (tt + tb);

    // Intersection / union / IoU
    float iw    = SS * (fminf(pl, tl) + fminf(pr, tr));
    float ih    = SS * (fminf(pt, tt) + fminf(pb, tb));
    float inter = fmaxf(iw, 0.0f) * fmaxf(ih, 0.0f);
    float uni   = fmaf(w1, h1, fmaf(w2, h2, -inter)) + EPSF;
    float iou   = inter * fast_rcp(uni);

    // Enclosing box diagonal
    float cw = SS * (fmaxf(pl, tl) + fmaxf(pr, tr));
    float ch = SS * (fmaxf(pt, tt) + fmaxf(pb, tb));
    float c2 = fmaf(cw, cw, ch * ch) + EPSF;

    // Center distance (centers cancel: only l/r, t/b asymmetries remain)
    float dx   = SS * ((tr - tl) - (pr - pl));
    float dy   = SS * ((tb - tt) - (pb - pt));
    float rho2 = 0.25f * fmaf(dx, dx, dy * dy);

    // v term: atan(w2/h2e) - atan(w1/h1e) == atan((w2*h1e - w1*h2e)/(h1e*h2e + w1*w2))
    float h1e  = h1 + EPSF, h2e = h2 + EPSF;
    float anum = fmaf(w2, h1e, -w1 * h2e);
    float aden = fmaf(h1e, h2e, w1 * w2);
    float dat  = fast_atan(anum * fast_rcp(aden));
    float v    = FOUR_PI2 * dat * dat;

    float alpha = v * fast_rcp(v - iou + (1.0f + EPSF));
    float ciou  = iou - rho2 * fast_rcp(c2) - alpha * v;

    s0 = fmaf(1.0f - ciou, m, s0);
    s1 += m;
}

__global__ void __launch_bounds__(NTHR)
ciou_partial_kernel(const float* __restrict__ pred,
                    const float* __restrict__ targ,
                    const unsigned char* __restrict__ mask,
                    float* __restrict__ ws) {
    float s0 = 0.0f, s1 = 0.0f;

    for (int i = blockIdx.x * NTHR + threadIdx.x; i < NVEC; i += NBLK * NTHR) {
        int b    = i / HW4;
        int p4   = i - b * HW4;
        int base = b * 4 * HW + p4 * 4;   // [B,4,H,W]: channel stride = HW
        int mbas = b * HW + p4 * 4;       // [B,1,H,W]

        // Non-temporal b128 streaming loads (single-use data: TH=NT)
        v4f pl = __builtin_nontemporal_load((const v4f*)(pred + base));
        v4f pt = __builtin_nontemporal_load((const v4f*)(pred + base + HW));
        v4f pr = __builtin_nontemporal_load((const v4f*)(pred + base + 2 * HW));
        v4f pb = __builtin_nontemporal_load((const v4f*)(pred + base + 3 * HW));
        v4f tl = __builtin_nontemporal_load((const v4f*)(targ + base));
        v4f tt = __builtin_nontemporal_load((const v4f*)(targ + base + HW));
        v4f tr = __builtin_nontemporal_load((const v4f*)(targ + base + 2 * HW));
        v4f tb = __builtin_nontemporal_load((const v4f*)(targ + base + 3 * HW));
        v4u8 mm = __builtin_nontemporal_load((const v4u8*)(mask + mbas));

        #pragma unroll
        for (int j = 0; j < 4; ++j) {
            float m = mm[j] ? 1.0f : 0.0f;
            ciou_accum(pl[j], pt[j], pr[j], pb[j],
                       tl[j], tt[j], tr[j], tb[j], m, s0, s1);
        }
    }

    // wave32 shuffle reduction
    #pragma unroll
    for (int off = 16; off > 0; off >>= 1) {
        s0 += __shfl_down(s0, off, 32);
        s1 += __shfl_down(s1, off, 32);
    }

    __shared__ float r0[NTHR / 32], r1[NTHR / 32];
    int wid = threadIdx.x >> 5, lid = threadIdx.x & 31;
    if (lid == 0) { r0[wid] = s0; r1[wid] = s1; }
    __syncthreads();
    if (threadIdx.x == 0) {
        float a = 0.0f, c = 0.0f;
        #pragma unroll
        for (int w = 0; w < NTHR / 32; ++w) { a += r0[w]; c += r1[w]; }
        ws[2 * blockIdx.x]     = a;
        ws[2 * blockIdx.x + 1] = c;
    }
}

// One wave (EXEC all ones, as WMMA requires). Per-lane partial pair is placed
// in the f32 16x4 A-matrix (lane m: a.x=A[m][0], a.y=A[m][1]; lane 16+m:
// a.x=A[m][2], a.y=A[m][3]); B (4x16) selects: column 0 sums the loss slots
// K={0,2}, column 1 sums the mask slots K={1,3}. Summing D's 8 VGPRs per lane
// then gives lanes {0,16} = loss row-halves, lanes {1,17} = mask row-halves.
__global__ void __launch_bounds__(32)
ciou_final_kernel(const float* __restrict__ ws, float* __restrict__ out) {
    int lane = threadIdx.x;
    float s0 = 0.0f, s1 = 0.0f;
    for (int i = lane; i < NBLK; i += 32) {
        s0 += ws[2 * i];
        s1 += ws[2 * i + 1];
    }

    float loss_sum, mask_sum;
#if __has_builtin(__builtin_amdgcn_wmma_f32_16x16x4_f32)
    v2f a; a.x = s0; a.y = s1;
    int l15 = lane & 15;
    v2f bsel;
    bsel.x = (l15 == 0) ? 1.0f : 0.0f;   // B[0][0]=B[2][0]=1 -> column 0: loss
    bsel.y = (l15 == 1) ? 1.0f : 0.0f;   // B[1][1]=B[3][1]=1 -> column 1: mask
    v8f c = {};
    c = __builtin_amdgcn_wmma_f32_16x16x4_f32(false, a, false, bsel,
                                              (short)0, c, false, false);
    float t = c[0] + c[1] + c[2] + c[3] + c[4] + c[5] + c[6] + c[7];
    loss_sum = __shfl(t, 0, 32) + __shfl(t, 16, 32);
    mask_sum = __shfl(t, 1, 32) + __shfl(t, 17, 32);
#else
    #pragma unroll
    for (int off = 16; off > 0; off >>= 1) {
        s0 += __shfl_down(s0, off, 32);
        s1 += __shfl_down(s1, off, 32);
    }
    loss_sum = __shfl(s0, 0, 32);
    mask_sum = __shfl(s1, 0, 32);
#endif

    if (lane == 0) out[0] = loss_sum / fmaxf(mask_sum, 1.0f);
}

extern "C" void kernel_launch(void* const* d_in, const int* in_sizes, int n_in,
                              void* d_out, int out_size, void* d_ws, size_t ws_size,
                              hipStream_t stream) {
    (void)in_sizes; (void)n_in; (void)out_size; (void)ws_size;
    const float*         pred = (const float*)d_in[0];
    const float*         targ = (const float*)d_in[1];
    const unsigned char* mask = (const unsigned char*)d_in[2]; // jnp.bool_: 1 byte/elem
    float* ws  = (float*)d_ws;   // needs NBLK*2*4 = 4 KB
    float* out = (float*)d_out;

    ciou_partial_kernel<<<NBLK, NTHR, 0, stream>>>(pred, targ, mask, ws);
    ciou_final_kernel<<<1, 32, 0, stream>>>(ws, out);
}